// PointConvCorrespondences_37546604101732
// MI455X (gfx1250) — compile-verified
//
#include <hip/hip_runtime.h>

typedef __attribute__((ext_vector_type(2))) float v2f;
typedef __attribute__((ext_vector_type(8))) float v8f;

#define BATCH 2
#define CH    3
#define DIM   64
#define KF    67          // D + 3 feature dims
#define KPAD  68          // padded to multiple of 4 (WMMA K-step)
#define KSTEP (KPAD / 4)  // 17 WMMA f32 16x16x4 steps
#define N1V   16384
#define N2V   16384
#define NTILE 64          // B columns staged in LDS per outer iteration
#define ROWS_PER_BLOCK 64
#define WAVES 4

// ---------------------------------------------------------------------------
// Kernel 1: |f2_n|^2 = sum over 67 dims (points2 then xyz2), per (b, n).
// ---------------------------------------------------------------------------
__global__ void norms2_kernel(const float* __restrict__ xyz2,
                              const float* __restrict__ points2,
                              float* __restrict__ norms) {
  int idx = blockIdx.x * blockDim.x + threadIdx.x;
  if (idx >= BATCH * N2V) return;
  int b = idx / N2V, n = idx - b * N2V;
  const float* p = points2 + (size_t)b * DIM * N2V + n;
  const float* x = xyz2 + (size_t)b * CH * N2V + n;
  float s = 0.f;
#pragma unroll 8
  for (int k = 0; k < DIM; ++k) { float v = p[(size_t)k * N2V]; s += v * v; }
#pragma unroll
  for (int c = 0; c < CH; ++c) { float v = x[(size_t)c * N2V]; s += v * v; }
  norms[idx] = s;
}

// ---------------------------------------------------------------------------
// Kernel 2: fused WMMA-f32 GEMM + argmin + gather.
// Block = 4 wave32 waves, 64 query rows; each wave owns a 16-row A slab
// kept in VGPRs; B tiles staged via LDS and shared by all waves.
// ---------------------------------------------------------------------------
__global__ __launch_bounds__(WAVES * 32)
void nn_argmin_kernel(const float* __restrict__ xyz1,
                      const float* __restrict__ xyz2,
                      const float* __restrict__ points1,
                      const float* __restrict__ points2,
                      const float* __restrict__ norms,
                      float* __restrict__ out) {
  __shared__ __align__(16) float ldsB[NTILE * KPAD];  // [col][k] (transposed)
  __shared__ float ldsN[NTILE];

  const int tid  = threadIdx.x;
  const int lane = tid & 31;
  const int wave = tid >> 5;
  const int half = lane >> 4;   // 0: lanes 0-15, 1: lanes 16-31
  const int l16  = lane & 15;
  const int b    = blockIdx.y;
  const int rowBase = blockIdx.x * ROWS_PER_BLOCK + wave * 16;
  const int m = rowBase + l16;  // this lane's A-matrix row (M = lane%16)

  const float* p1  = points1 + (size_t)b * DIM * N1V;
  const float* x1  = xyz1    + (size_t)b * CH  * N1V;
  const float* p2  = points2 + (size_t)b * DIM * N2V;
  const float* x2  = xyz2    + (size_t)b * CH  * N2V;
  const float* nrm = norms   + (size_t)b * N2V;

  // Preload A slab (16 rows x KPAD) into 34 VGPRs; reused for all N2 tiles.
  // f32 WMMA A layout: lanes 0-15 hold K = 4s+{0,1}, lanes 16-31 K = 4s+{2,3}.
  v2f A[KSTEP];
#pragma unroll
  for (int s = 0; s < KSTEP; ++s) {
#pragma unroll
    for (int j = 0; j < 2; ++j) {
      int k = 4 * s + 2 * half + j;
      float v;
      if (k < DIM)     v = p1[(size_t)k * N1V + m];
      else if (k < KF) v = x1[(size_t)(k - DIM) * N1V + m];
      else             v = 0.f;   // K padding
      A[s][j] = v;
    }
  }

  float minv[8];
  int   mini[8];
#pragma unroll
  for (int r = 0; r < 8; ++r) { minv[r] = __builtin_inff(); mini[r] = 0; }

  for (int n0 = 0; n0 < N2V; n0 += NTILE) {
    __syncthreads();
    // Cooperative stage: ldsB[col][k] = f2[k][n0+col] (coalesced global reads).
    for (int l = tid; l < KPAD * NTILE; l += WAVES * 32) {
      int k   = l >> 6;     // 0..67
      int col = l & 63;
      float v = 0.f;
      if (k < DIM)     v = p2[(size_t)k * N2V + n0 + col];
      else if (k < KF) v = x2[(size_t)(k - DIM) * N2V + n0 + col];
      ldsB[col * KPAD + k] = v;
    }
    if (tid < NTILE) ldsN[tid] = nrm[n0 + tid];
    __syncthreads();

#pragma unroll
    for (int sub = 0; sub < NTILE / 16; ++sub) {
      const int col = sub * 16 + l16;
      v8f acc = {};
      // 17 chained f32 WMMAs: acc = A(16x68) * B(68x16)
#pragma unroll
      for (int s = 0; s < KSTEP; ++s) {
        // B layout mirrors A: lane's K-pair (4s+2*half, 4s+2*half+1) is
        // contiguous in the transposed LDS tile -> one 64-bit ds load.
        v2f Bv = *(const v2f*)&ldsB[col * KPAD + 4 * s + 2 * half];
        acc = __builtin_amdgcn_wmma_f32_16x16x4_f32(
            false, A[s], false, Bv, (short)0, acc, false, false);
      }
      // Epilogue: cand = |b_n|^2 - 2*(a.b); |a|^2 drops out of argmin.
      const float bn  = ldsN[col];
      const int nidx  = n0 + col;
#pragma unroll
      for (int r = 0; r < 8; ++r) {
        float cand = bn - 2.0f * acc[r];
        if (cand < minv[r]) { minv[r] = cand; mini[r] = nidx; }
      }
    }
  }

  // Cross-lane argmin over the 16 N-partitions (xor masks < 16 stay within
  // each half-wave, matching the C/D layout: VGPR r -> M = r + 8*half).
  float* corres1 = out;
  float* corres2 = out + (size_t)BATCH * N1V;
  float* dir     = out + (size_t)2 * BATCH * N1V;
#pragma unroll
  for (int r = 0; r < 8; ++r) {
    float v  = minv[r];
    int   ix = mini[r];
#pragma unroll
    for (int d = 1; d < 16; d <<= 1) {
      float ov = __shfl_xor(v, d, 32);
      int   oi = __shfl_xor(ix, d, 32);
      if (ov < v || (ov == v && oi < ix)) { v = ov; ix = oi; }
    }
    if (l16 == 0) {
      int row = rowBase + r + 8 * half;
      corres1[(size_t)b * N1V + row] = (float)row;
      corres2[(size_t)b * N1V + row] = (float)ix;
#pragma unroll
      for (int c = 0; c < CH; ++c) {
        dir[((size_t)b * CH + c) * N1V + row] =
            x2[(size_t)c * N2V + ix] - x1[(size_t)c * N1V + row];
      }
    }
  }
}

// ---------------------------------------------------------------------------
extern "C" void kernel_launch(void* const* d_in, const int* in_sizes, int n_in,
                              void* d_out, int out_size, void* d_ws, size_t ws_size,
                              hipStream_t stream) {
  const float* xyz1    = (const float*)d_in[0];
  const float* xyz2    = (const float*)d_in[1];
  const float* points1 = (const float*)d_in[2];
  const float* points2 = (const float*)d_in[3];
  float* out   = (float*)d_out;
  float* norms = (float*)d_ws;  // BATCH * N2 floats = 128 KB scratch

  dim3 g1((BATCH * N2V + 255) / 256), b1(256);
  norms2_kernel<<<g1, b1, 0, stream>>>(xyz2, points2, norms);

  dim3 g2(N1V / ROWS_PER_BLOCK, BATCH), b2(WAVES * 32);
  nn_argmin_kernel<<<g2, b2, 0, stream>>>(xyz1, xyz2, points1, points2, norms, out);
}